// GaussianMixLatentSpace_51041391346012
// MI455X (gfx1250) — compile-verified
//
#include <hip/hip_runtime.h>
#include <math.h>

typedef __attribute__((ext_vector_type(2))) float v2f;
typedef __attribute__((ext_vector_type(8))) float v8f;

#define N_      8192
#define K_      64
#define M_      100
#define D_      16
#define COLS    (K_ * M_)              // 6400 = 400 tiles of 16
#define TILES_PER_WAVE 50              // 400 tiles / 8 waves
#define SIGMA_INV_LOG2E 14.4269504088896341f   // 10 * log2(e)

// ---------------- prep 1: cholesky + transform + g2 + log2-prop ----------------
__global__ __launch_bounds__(128) void prep_transform(
        const float* __restrict__ points, const float* __restrict__ mu,
        const float* __restrict__ cov,    const float* __restrict__ prop,
        float* __restrict__ Gf, float* __restrict__ g2, float* __restrict__ logp)
{
    __shared__ float Cv[16][16];
    __shared__ float Ls[16][16];
    __shared__ float Gs[M_][16];
    const int k = blockIdx.x;
    const int t = threadIdx.x;

    for (int idx = t; idx < 256; idx += 128)
        Cv[idx >> 4][idx & 15] = cov[k * 256 + idx];
    __syncthreads();

    if (t == 0) {   // 16x16 Cholesky, lower triangle (tiny; precise math fine)
        for (int c = 0; c < 16; ++c) {
            float s = Cv[c][c];
            for (int j = 0; j < c; ++j) s -= Ls[c][j] * Ls[c][j];
            float dc = sqrtf(fmaxf(s, 1e-20f));
            Ls[c][c] = dc;
            float inv = 1.0f / dc;
            for (int r = c + 1; r < 16; ++r) {
                float v = Cv[r][c];
                for (int j = 0; j < c; ++j) v -= Ls[r][j] * Ls[c][j];
                Ls[r][c] = v * inv;
            }
        }
    }
    __syncthreads();

    // G[k,j,:] = points[k,j,:] @ L^T + mu[k,:]
    for (int idx = t; idx < M_ * 16; idx += 128) {
        int j = idx >> 4, e = idx & 15;
        const float* pr = points + (size_t)(k * M_ + j) * 16;
        float s = mu[k * 16 + e];
        for (int q = 0; q <= e; ++q) s += pr[q] * Ls[e][q];
        Gs[j][e] = s;
        Gf[(size_t)(k * M_ + j) * 16 + e] = s;
    }
    __syncthreads();

    float lp2 = log2f(prop[k]);        // log2 domain for native v_exp_f32 softmax
    for (int j = t; j < M_; j += 128) {
        float s = 0.f;
        for (int e = 0; e < 16; ++e) { float v = Gs[j][e]; s += v * v; }
        g2[k * M_ + j]   = s;
        logp[k * M_ + j] = lp2;
    }
}

// ---------------- prep 2: row norms of z ----------------
__global__ __launch_bounds__(256) void prep_z2(const float* __restrict__ z,
                                               float* __restrict__ z2)
{
    int r = blockIdx.x * 256 + threadIdx.x;
    if (r < N_) {
        const float* p = z + (size_t)r * 16;
        float s = 0.f;
        for (int e = 0; e < 16; ++e) s += p[e] * p[e];
        z2[r] = s;
    }
}

__device__ __forceinline__ void load_tile(const float* __restrict__ Gf,
                                          const float* __restrict__ g2,
                                          const float* __restrict__ logp,
                                          int col, int hi,
                                          v2f b[4], float& g2c, float& lpc)
{
    const float2* br = (const float2*)(Gf + (size_t)col * 16);
    #pragma unroll
    for (int s = 0; s < 4; ++s) {
        float2 t2 = br[2 * s + hi];
        b[s].x = t2.x; b[s].y = t2.y;
    }
    g2c = g2[col];
    lpc = logp[col];
}

// ---------------- main: WMMA matmul + fused two-pass softmax ----------------
__global__ __launch_bounds__(256) void gm_main(
        const float* __restrict__ z,  const float* __restrict__ Gf,
        const float* __restrict__ g2, const float* __restrict__ logp,
        const float* __restrict__ z2,
        float* __restrict__ point_resp, float* __restrict__ comp_resp)
{
    __shared__ float wM[16][8];
    __shared__ float wS[16][8];
    __shared__ float rowM[16];
    __shared__ float rowS[16];
    __shared__ float compS[16][K_];

    const int tid   = threadIdx.x;
    const int wid   = tid >> 5;          // 8 waves; wave owns cols [wid*800, wid*800+800)
    const int lane  = tid & 31;          // wave32
    const int lhalf = lane & 15;
    const int hi    = lane >> 4;
    const int voff  = hi * 8;            // D rows covered: voff..voff+7
    const int koff  = hi * 2;            // A/B K offset per ISA layout
    const int rowBase = blockIdx.x * 16;
    const int tile0   = wid * TILES_PER_WAVE;

    for (int i = tid; i < 16 * K_; i += 256) ((float*)compS)[i] = 0.f;

    // A fragment (z tile): lane = M, VGPR pair holds K = 4s+koff, 4s+koff+1
    v2f a[4];
    {
        const float* zr = z + (size_t)(rowBase + lhalf) * 16;
        #pragma unroll
        for (int s = 0; s < 4; ++s) {
            a[s].x = zr[4 * s + koff];
            a[s].y = zr[4 * s + koff + 1];
        }
    }
    float z2l[8];
    #pragma unroll
    for (int i = 0; i < 8; ++i) z2l[i] = z2[rowBase + voff + i];

    float runM[8], runS[8];
    #pragma unroll
    for (int i = 0; i < 8; ++i) { runM[i] = -3.0e38f; runS[i] = 0.f; }

    // ---- pass 1: online row max / sum-of-exp2 (no global spill) ----
    v2f bufB[4]; float bufG2, bufLP;
    load_tile(Gf, g2, logp, tile0 * 16 + lhalf, hi, bufB, bufG2, bufLP);
    for (int tt = 0; tt < TILES_PER_WAVE; ++tt) {
        v2f b[4];
        #pragma unroll
        for (int s = 0; s < 4; ++s) b[s] = bufB[s];
        const float g2c = bufG2, lpc = bufLP;
        if (tt + 1 < TILES_PER_WAVE)      // double-buffer next tile
            load_tile(Gf, g2, logp, (tile0 + tt + 1) * 16 + lhalf, hi, bufB, bufG2, bufLP);

        v8f acc = {};
        #pragma unroll
        for (int s = 0; s < 4; ++s)
            acc = __builtin_amdgcn_wmma_f32_16x16x4_f32(
                      false, a[s], false, b[s], (short)0, acc, false, false);
        #pragma unroll
        for (int i = 0; i < 8; ++i) {
            float d2    = fmaxf(z2l[i] + g2c - 2.0f * acc[i], 0.0f);
            float logit = lpc - __builtin_amdgcn_sqrtf(d2) * SIGMA_INV_LOG2E;
            float nm    = fmaxf(runM[i], logit);
            runS[i] = runS[i] * __builtin_amdgcn_exp2f(runM[i] - nm)
                    + __builtin_amdgcn_exp2f(logit - nm);
            runM[i] = nm;
        }
    }
    // butterfly merge within each 16-lane half (wave32 shuffles)
    #pragma unroll
    for (int mask = 1; mask < 16; mask <<= 1) {
        #pragma unroll
        for (int i = 0; i < 8; ++i) {
            float om = __shfl_xor(runM[i], mask, 32);
            float os = __shfl_xor(runS[i], mask, 32);
            float nm = fmaxf(runM[i], om);
            runS[i] = runS[i] * __builtin_amdgcn_exp2f(runM[i] - nm)
                    + os * __builtin_amdgcn_exp2f(om - nm);
            runM[i] = nm;
        }
    }
    if (lhalf == 0)
        #pragma unroll
        for (int i = 0; i < 8; ++i) { wM[voff + i][wid] = runM[i]; wS[voff + i][wid] = runS[i]; }
    __syncthreads();
    if (tid < 16) {   // merge across the 8 waves
        float Mm = -3.0e38f, Ss = 0.f;
        for (int w = 0; w < 8; ++w) {
            float om = wM[tid][w], os = wS[tid][w];
            float nm = fmaxf(Mm, om);
            Ss = Ss * __builtin_amdgcn_exp2f(Mm - nm) + os * __builtin_amdgcn_exp2f(om - nm);
            Mm = nm;
        }
        rowM[tid] = Mm; rowS[tid] = Ss;
    }
    __syncthreads();

    float Mf[8], Sinv[8];
    #pragma unroll
    for (int i = 0; i < 8; ++i) {
        Mf[i]   = rowM[voff + i];
        Sinv[i] = 1.0f / rowS[voff + i];
    }

    // ---- pass 2: recompute (L2-resident inputs), normalize, NT-store once ----
    float acc8[8];
    #pragma unroll
    for (int i = 0; i < 8; ++i) acc8[i] = 0.f;
    int kcur = (tile0 * 16) / M_;        // wave's 800-col stretch = components kcur..kcur+7

    auto flush = [&](int kc_) {
        #pragma unroll
        for (int i = 0; i < 8; ++i) {
            float s = acc8[i];
            s += __shfl_xor(s, 1, 32);
            s += __shfl_xor(s, 2, 32);
            s += __shfl_xor(s, 4, 32);
            s += __shfl_xor(s, 8, 32);
            if (lhalf == 0) atomicAdd(&compS[voff + i][kc_], s);
            acc8[i] = 0.f;
        }
    };

    load_tile(Gf, g2, logp, tile0 * 16 + lhalf, hi, bufB, bufG2, bufLP);
    for (int tt = 0; tt < TILES_PER_WAVE; ++tt) {
        const int c0  = (tile0 + tt) * 16;
        const int col = c0 + lhalf;
        v2f b[4];
        #pragma unroll
        for (int s = 0; s < 4; ++s) b[s] = bufB[s];
        const float g2c = bufG2, lpc = bufLP;
        if (tt + 1 < TILES_PER_WAVE)
            load_tile(Gf, g2, logp, c0 + 16 + lhalf, hi, bufB, bufG2, bufLP);

        v8f acc = {};
        #pragma unroll
        for (int s = 0; s < 4; ++s)
            acc = __builtin_amdgcn_wmma_f32_16x16x4_f32(
                      false, a[s], false, b[s], (short)0, acc, false, false);

        const int kf = c0 / M_;          // dominant component of this tile (wave-uniform)
        if (kf != kcur) { flush(kcur); kcur = kf; }
        const int kc = col / M_;         // per-lane component (differs only at straddle)
        const bool inKf = (kc == kf);

        float p[8];
        #pragma unroll
        for (int i = 0; i < 8; ++i) {
            float d2    = fmaxf(z2l[i] + g2c - 2.0f * acc[i], 0.0f);
            float logit = lpc - __builtin_amdgcn_sqrtf(d2) * SIGMA_INV_LOG2E;
            p[i] = __builtin_amdgcn_exp2f(logit - Mf[i]) * Sinv[i];
            __builtin_nontemporal_store(p[i],
                &point_resp[(size_t)(rowBase + voff + i) * COLS + col]);
            if (inKf) acc8[i] += p[i];
        }
        if (!inKf) {                     // straddle lanes: direct LDS atomics (rare)
            #pragma unroll
            for (int i = 0; i < 8; ++i) atomicAdd(&compS[voff + i][kc], p[i]);
        }
    }
    flush(kcur);
    __syncthreads();
    for (int i = tid; i < 16 * K_; i += 256)
        comp_resp[(size_t)rowBase * K_ + i] = ((float*)compS)[i];
}

// ---------------- launch ----------------
extern "C" void kernel_launch(void* const* d_in, const int* in_sizes, int n_in,
                              void* d_out, int out_size, void* d_ws, size_t ws_size,
                              hipStream_t stream)
{
    const float* z      = (const float*)d_in[0];
    const float* points = (const float*)d_in[1];
    const float* mu     = (const float*)d_in[2];
    const float* cov    = (const float*)d_in[3];
    const float* prop   = (const float*)d_in[4];

    char*  ws   = (char*)d_ws;
    float* Gf   = (float*)(ws);                 // 6400*16 f32 = 409600 B
    float* g2   = (float*)(ws + 409600);        // 6400 f32
    float* logp = (float*)(ws + 435200);        // 6400 f32 (log2 domain)
    float* z2   = (float*)(ws + 460800);        // 8192 f32

    float* point_resp = (float*)d_out;
    float* comp_resp  = point_resp + (size_t)N_ * COLS;

    prep_transform<<<K_, 128, 0, stream>>>(points, mu, cov, prop, Gf, g2, logp);
    prep_z2<<<N_ / 256, 256, 0, stream>>>(z, z2);
    gm_main<<<N_ / 16, 256, 0, stream>>>(z, Gf, g2, logp, z2, point_resp, comp_resp);
}